// Wavelet_10720238370912
// MI455X (gfx1250) — compile-verified
//
#include <hip/hip_runtime.h>
#include <hip/hip_bf16.h>

#define B_DIM   1024
#define IN_DIM  512
#define OUT_DIM 512

typedef float v2f __attribute__((ext_vector_type(2)));
typedef float v8f __attribute__((ext_vector_type(8)));

// phi(z) = -z * exp(-z^2/2) = -z * exp2(z^2 * (-0.5*log2(e)))
__device__ __forceinline__ float wavelet_phi(float xv, float t, float sinv) {
    float z = (xv - t) * sinv;
    float e = __builtin_amdgcn_exp2f(z * z * -0.72134752044448169136f);
    return -z * e;
}

// Kernel 1: wavelet "generalized GEMM" -> ws[o][b] (o-major for coalesced BN reduce)
// grid = (OUT/16, B/128), block = 256 (8 waves). Each wave owns a 16b x 16o tile.
__launch_bounds__(256)
__global__ void wavelet_gemm_kernel(const float* __restrict__ x,
                                    const float* __restrict__ scale,
                                    const float* __restrict__ translation,
                                    const float* __restrict__ wweights,
                                    float* __restrict__ wsout /* [OUT][B] */)
{
    // LDS: translation tile + reciprocal-scale tile, 16 x 512 each = 64KB
    __shared__ float lds[2 * 16 * IN_DIM];
    float* t_lds = lds;
    float* s_lds = lds + 16 * IN_DIM;

    const int o0 = blockIdx.x * 16;

    // Cooperative stage: 2 matrices * 16 rows * 512 = 4096 float4, 16 per thread.
    for (int k = 0; k < 16; ++k) {
        int idx = threadIdx.x + k * 256;      // float4 index in [0, 4096)
        int m   = idx >> 11;                  // 0 = translation, 1 = scale
        int rem = idx & 2047;
        int oo  = rem >> 7;                   // row 0..15
        int ii  = (rem & 127) << 2;           // col 0..508 step 4
        const float* src = (m == 0) ? translation : scale;
        float4 v = *(const float4*)(src + (size_t)(o0 + oo) * IN_DIM + ii);
        if (m == 1) {                         // store 1/scale so phi uses a mul
            v.x = __builtin_amdgcn_rcpf(v.x);
            v.y = __builtin_amdgcn_rcpf(v.y);
            v.z = __builtin_amdgcn_rcpf(v.z);
            v.w = __builtin_amdgcn_rcpf(v.w);
        }
        *(float4*)(lds + m * (16 * IN_DIM) + oo * IN_DIM + ii) = v;
    }
    __syncthreads();

    const int lane = threadIdx.x & 31;
    const int wv   = threadIdx.x >> 5;
    const int n    = lane & 15;               // A row (b) this lane feeds; also C/D column
    const int hi   = lane >> 4;               // 0 or 1
    const int koff = hi * 2;                  // K pair (0,1) or (2,3) per 32-bit A/B layout
    const int b0   = blockIdx.y * 128 + wv * 16;

    // i-invariant selector masks and o-locals for the 4 o-groups
    bool sel0[4], sel1[4];
    int  oA0[4], oA1[4];
#pragma unroll
    for (int g = 0; g < 4; ++g) {
        int og = g * 4;
        oA0[g] = og + koff;
        oA1[g] = og + koff + 1;
        sel0[g] = (n == oA0[g]);
        sel1[g] = (n == oA1[g]);
    }

    v8f acc[4];
#pragma unroll
    for (int g = 0; g < 4; ++g) acc[g] = (v8f)0.0f;

    const float* xrow = x + (size_t)(b0 + n) * IN_DIM;          // A row M = lane&15
    const float* wrow = wweights + (size_t)(o0 + n) * IN_DIM;   // this lane's B column weights

    for (int i = 0; i < IN_DIM; i += 2) {
        v2f xv = *(const v2f*)(xrow + i);
        v2f wp = *(const v2f*)(wrow + i);
#pragma unroll
        for (int g = 0; g < 4; ++g) {
            v2f t0 = *(const v2f*)(t_lds + oA0[g] * IN_DIM + i);
            v2f s0 = *(const v2f*)(s_lds + oA0[g] * IN_DIM + i);
            v2f t1 = *(const v2f*)(t_lds + oA1[g] * IN_DIM + i);
            v2f s1 = *(const v2f*)(s_lds + oA1[g] * IN_DIM + i);

            v2f a_i0, a_i1;                    // A[b, K=koff], A[b, K=koff+1]
            a_i0.x = wavelet_phi(xv.x, t0.x, s0.x);
            a_i0.y = wavelet_phi(xv.x, t1.x, s1.x);
            a_i1.x = wavelet_phi(xv.y, t0.y, s0.y);
            a_i1.y = wavelet_phi(xv.y, t1.y, s1.y);

            v2f b_i0, b_i1;                    // block-diagonal selector weights
            b_i0.x = sel0[g] ? wp.x : 0.0f;
            b_i0.y = sel1[g] ? wp.x : 0.0f;
            b_i1.x = sel0[g] ? wp.y : 0.0f;
            b_i1.y = sel1[g] ? wp.y : 0.0f;

            acc[g] = __builtin_amdgcn_wmma_f32_16x16x4_f32(
                         false, a_i0, false, b_i0, (short)0, acc[g], false, false);
            acc[g] = __builtin_amdgcn_wmma_f32_16x16x4_f32(
                         false, a_i1, false, b_i1, (short)0, acc[g], false, false);
        }
    }

    v8f d = acc[0] + acc[1] + acc[2] + acc[3];
    // C/D layout: lane column N = lane&15, rows M = r + 8*hi
    float* outcol = wsout + (size_t)(o0 + n) * B_DIM + b0 + hi * 8;
#pragma unroll
    for (int r = 0; r < 8; ++r) outcol[r] = d[r];
}

// Kernel 2: per-channel BatchNorm (training mode, biased var), one block per o.
__launch_bounds__(256)
__global__ void bn_kernel(const float* __restrict__ wsout, // [OUT][B]
                          const float* __restrict__ gamma,
                          const float* __restrict__ beta,
                          float* __restrict__ out /* [B][OUT] */)
{
    __shared__ float r1[256];
    __shared__ float r2[256];
    const int o = blockIdx.x;
    const int t = threadIdx.x;

    float v[4];
    float s = 0.0f, sq = 0.0f;
#pragma unroll
    for (int r = 0; r < 4; ++r) {
        v[r] = wsout[(size_t)o * B_DIM + t + r * 256];   // coalesced column read
        s  += v[r];
        sq += v[r] * v[r];
    }
    r1[t] = s; r2[t] = sq;
    __syncthreads();
    for (int off = 128; off > 0; off >>= 1) {
        if (t < off) { r1[t] += r1[t + off]; r2[t] += r2[t + off]; }
        __syncthreads();
    }
    float mean = r1[0] * (1.0f / B_DIM);
    float var  = r2[0] * (1.0f / B_DIM) - mean * mean;
    float istd = __builtin_amdgcn_rsqf(var + 1e-5f);
    float g = gamma[o], bb = beta[o];
#pragma unroll
    for (int r = 0; r < 4; ++r)
        out[(size_t)(t + r * 256) * OUT_DIM + o] = g * (v[r] - mean) * istd + bb;
}

extern "C" void kernel_launch(void* const* d_in, const int* in_sizes, int n_in,
                              void* d_out, int out_size, void* d_ws, size_t ws_size,
                              hipStream_t stream) {
    const float* x           = (const float*)d_in[0];
    const float* scale       = (const float*)d_in[1];
    const float* translation = (const float*)d_in[2];
    const float* wweights    = (const float*)d_in[3];
    const float* gamma       = (const float*)d_in[4];
    const float* beta        = (const float*)d_in[5];
    float* ws  = (float*)d_ws;          // 2MB: wavelet_output in [OUT][B] layout
    float* out = (float*)d_out;

    dim3 g1(OUT_DIM / 16, B_DIM / 128);
    wavelet_gemm_kernel<<<g1, 256, 0, stream>>>(x, scale, translation, wweights, ws);
    bn_kernel<<<OUT_DIM, 256, 0, stream>>>(ws, gamma, beta, out);
}